// GraphClassifier_44238163149209
// MI455X (gfx1250) — compile-verified
//
#include <hip/hip_runtime.h>

typedef float v2f __attribute__((ext_vector_type(2)));
typedef float v8f __attribute__((ext_vector_type(8)));

#define IN_CH   128
#define HID_CH  128
#define OUT_CH  64
#define N_GRAPHS 64

// ---------------------------------------------------------------------------
// deg[n] = 1 (self loop); also zero the pooled output
// ---------------------------------------------------------------------------
__global__ void init_kernel(float* __restrict__ deg, float* __restrict__ out,
                            int n, int out_n) {
    int i = blockIdx.x * blockDim.x + threadIdx.x;
    if (i < n)     deg[i] = 1.0f;
    if (i < out_n) out[i] = 0.0f;
}

__global__ void deg_edges_kernel(const long long* __restrict__ dst,
                                 float* __restrict__ deg, int E) {
    int e = blockIdx.x * blockDim.x + threadIdx.x;
    if (e < E) atomicAdd(&deg[(int)dst[e]], 1.0f);
}

__global__ void dinv_kernel(float* __restrict__ deg, int n) {
    int i = blockIdx.x * blockDim.x + threadIdx.x;
    if (i < n) deg[i] = rsqrtf(deg[i]);   // deg >= 1 always (self loop)
}

// ---------------------------------------------------------------------------
// Row-panel GEMM:  HS = AGG = dinv ⊙ (X @ W)  via V_WMMA_F32_16X16X4_F32.
// One wave computes a 16-row × (NT*16)-col panel: the A fragment (16x4) is
// loaded ONCE per k-step and reused by NT consecutive WMMAs, so X is read
// exactly once from global. W (<=64KB) is cache-resident.
//   A 16x4 layout : lanes 0-15 rows M=0..15 {K,K+1}; lanes 16-31 {K+2,K+3}
//   B 4x16 layout : lanes 0-15 cols N=0..15 {K,K+1}; lanes 16-31 {K+2,K+3}
//   C/D           : VGPR r -> M=r (lanes 0-15, N=lane) / M=r+8 (lanes 16-31)
// ---------------------------------------------------------------------------
template <int NT>   // number of 16-wide N tiles: Ncols = NT*16
__global__ __launch_bounds__(128)
void gemm_wmma_panel(const float* __restrict__ X, const float* __restrict__ W,
                     const float* __restrict__ dinv,
                     float* __restrict__ HS, float* __restrict__ AGG,
                     int M, int K) {
    constexpr int Ncols = NT * 16;
    int wave = threadIdx.x >> 5;
    int lane = threadIdx.x & 31;
    int tm   = blockIdx.x * 4 + wave;        // 16-row panel index
    if (tm >= (M >> 4)) return;              // wave-uniform; EXEC stays all-1s
    int half = lane >> 4;                    // 0: lanes 0-15, 1: lanes 16-31
    int l16  = lane & 15;

    v8f c[NT];
#pragma unroll
    for (int t = 0; t < NT; ++t) c[t] = v8f{0.f,0.f,0.f,0.f,0.f,0.f,0.f,0.f};

    const float* xrow = X + (size_t)(tm * 16 + l16) * K;  // this lane's A row

    for (int k = 0; k < K; k += 4) {
        int ka = k + 2 * half;
        v2f a;
        a.x = xrow[ka];
        a.y = xrow[ka + 1];
        const float* w0 = W + (size_t)ka * Ncols + l16;
#pragma unroll
        for (int t = 0; t < NT; ++t) {
            v2f b;
            b.x = w0[t * 16];
            b.y = w0[t * 16 + Ncols];
            c[t] = __builtin_amdgcn_wmma_f32_16x16x4_f32(
                       /*neg_a=*/false, a, /*neg_b=*/false, b,
                       /*c_mod=*/(short)0, c[t],
                       /*reuse_a=*/false, /*reuse_b=*/false);
        }
    }

    // epilogue: scale row m by dinv[m], write gather source + seeded accumulator
    float dv[8];
#pragma unroll
    for (int r = 0; r < 8; ++r) dv[r] = dinv[tm * 16 + r + half * 8];

#pragma unroll
    for (int t = 0; t < NT; ++t) {
#pragma unroll
        for (int r = 0; r < 8; ++r) {
            int gm = tm * 16 + r + half * 8;
            int gn = t * 16 + l16;
            float v = c[t][r] * dv[r];
            size_t o = (size_t)gm * Ncols + gn;
            HS[o]  = v;                      // gather source for edges
            AGG[o] = v;                      // accumulator seeded w/ self-loop
        }
    }
}

// ---------------------------------------------------------------------------
// AGG[dst] += HS[src]   (float4 gather, 4x f32 atomic scatter per thread)
// ---------------------------------------------------------------------------
__global__ void edge_scatter_kernel(const long long* __restrict__ src,
                                    const long long* __restrict__ dst,
                                    const float* __restrict__ HS,
                                    float* __restrict__ AGG, int E, int C) {
    int idx = blockIdx.x * blockDim.x + threadIdx.x;
    int q4  = C >> 2;
    int e   = idx / q4;
    if (e >= E) return;
    int q = idx - e * q4;
    int s = (int)src[e];
    int d = (int)dst[e];
    float4 v = ((const float4*)(HS + (size_t)s * C))[q];
    float* o = AGG + (size_t)d * C + q * 4;
    atomicAdd(o + 0, v.x);
    atomicAdd(o + 1, v.y);
    atomicAdd(o + 2, v.z);
    atomicAdd(o + 3, v.w);
}

// ---------------------------------------------------------------------------
// X2 = relu(dinv ⊙ AGG + b)
// ---------------------------------------------------------------------------
__global__ void finalize_relu_kernel(const float* __restrict__ AGG,
                                     const float* __restrict__ dinv,
                                     const float* __restrict__ bias,
                                     float* __restrict__ Xout, int n, int C) {
    int idx = blockIdx.x * blockDim.x + threadIdx.x;
    if (idx >= n * C) return;
    int node = idx / C;
    int ch   = idx - node * C;
    float v = dinv[node] * AGG[idx] + bias[ch];
    Xout[idx] = fmaxf(v, 0.0f);
}

// ---------------------------------------------------------------------------
// out[batch[n]] += dinv[n] ⊙ AGG2[n] + b2   (finalize layer 2 + add-pool)
// ---------------------------------------------------------------------------
__global__ void pool_kernel(const float* __restrict__ AGG2,
                            const float* __restrict__ dinv,
                            const float* __restrict__ b2,
                            const long long* __restrict__ batch,
                            float* __restrict__ out, int n) {
    int idx = blockIdx.x * blockDim.x + threadIdx.x;
    if (idx >= n * OUT_CH) return;
    int node = idx / OUT_CH;
    int ch   = idx - node * OUT_CH;
    float v = dinv[node] * AGG2[idx] + b2[ch];
    atomicAdd(&out[(int)batch[node] * OUT_CH + ch], v);
}

// ---------------------------------------------------------------------------
extern "C" void kernel_launch(void* const* d_in, const int* in_sizes, int n_in,
                              void* d_out, int out_size, void* d_ws, size_t ws_size,
                              hipStream_t stream) {
    const float*     x     = (const float*)d_in[0];
    const long long* edge  = (const long long*)d_in[1];   // int64 [2, E]
    const long long* batch = (const long long*)d_in[2];   // int64 [N], sorted
    const float*     W1    = (const float*)d_in[3];
    const float*     b1    = (const float*)d_in[4];
    const float*     W2    = (const float*)d_in[5];
    const float*     b2    = (const float*)d_in[6];
    float*           out   = (float*)d_out;

    int N = in_sizes[2];
    int E = in_sizes[1] / 2;
    const long long* src = edge;
    const long long* dst = edge + E;

    // workspace layout: deg/dinv | bufA (HS1 -> X2) | bufB (AGG1 -> HS2|AGG2)
    char* ws = (char*)d_ws;
    size_t off = 0;
    float* deg  = (float*)(ws + off); off += (((size_t)N * 4) + 255) & ~(size_t)255;
    float* bufA = (float*)(ws + off); off += (size_t)N * HID_CH * 4;
    float* bufB = (float*)(ws + off);
    float* HS1  = bufA;
    float* AGG1 = bufB;
    float* X2   = bufA;                       // overwrites HS1 (done with it)
    float* HS2  = bufB;                       // overwrites AGG1 (done with it)
    float* AGG2 = bufB + (size_t)N * OUT_CH;

    // degree / dinv
    int mx = N > N_GRAPHS * OUT_CH ? N : N_GRAPHS * OUT_CH;
    init_kernel<<<(mx + 255) / 256, 256, 0, stream>>>(deg, out, N, N_GRAPHS * OUT_CH);
    deg_edges_kernel<<<(E + 255) / 256, 256, 0, stream>>>(dst, deg, E);
    dinv_kernel<<<(N + 255) / 256, 256, 0, stream>>>(deg, N);

    int panels = N / 16;                      // 50000/16 = 3125, exact

    // ---- layer 1: h = relu(GCNConv(x; W1, b1)) ----
    gemm_wmma_panel<HID_CH / 16><<<(panels + 3) / 4, 128, 0, stream>>>(
        x, W1, deg, HS1, AGG1, N, IN_CH);
    edge_scatter_kernel<<<(E * (HID_CH / 4) + 255) / 256, 256, 0, stream>>>(
        src, dst, HS1, AGG1, E, HID_CH);
    finalize_relu_kernel<<<(N * HID_CH + 255) / 256, 256, 0, stream>>>(
        AGG1, deg, b1, X2, N, HID_CH);

    // ---- layer 2: h = GCNConv(h; W2, b2), then global add pool ----
    gemm_wmma_panel<OUT_CH / 16><<<(panels + 3) / 4, 128, 0, stream>>>(
        X2, W2, deg, HS2, AGG2, N, HID_CH);
    edge_scatter_kernel<<<(E * (OUT_CH / 4) + 255) / 256, 256, 0, stream>>>(
        src, dst, HS2, AGG2, E, OUT_CH);
    pool_kernel<<<(N * OUT_CH + 255) / 256, 256, 0, stream>>>(
        AGG2, deg, b2, batch, out, N);
}